// GaussianMixtureModel_91070486544503
// MI455X (gfx1250) — compile-verified
//
#include <hip/hip_runtime.h>
#include <math.h>

typedef __attribute__((ext_vector_type(2))) float v2f;
typedef __attribute__((ext_vector_type(8))) float v8f;

#define LOG_2PI_F 1.8378770664093453f
#define B_SZ 32768
#define K_SZ 64
#define D_SZ 64

// ---------------------------------------------------------------------------
// Kernel 1: per-component constant ck[k] = logw[k] - half_log_det[k] - 32*log(2pi)
// logw = log_softmax(weights_raw / 8); half_log_det = sum_i scale_raw[k,i,i]/512
// ---------------------------------------------------------------------------
__global__ __launch_bounds__(64) void prep_wh(const float* __restrict__ wr,
                                              const float* __restrict__ sr,
                                              float* __restrict__ ck_out) {
    __shared__ float sh[64];
    const int k = threadIdx.x;
    const float w = wr[k] * 0.125f;
    sh[k] = w;
    __syncthreads();
    float mx = -INFINITY;
    #pragma unroll 8
    for (int i = 0; i < 64; ++i) mx = fmaxf(mx, sh[i]);
    float sum = 0.0f;
    #pragma unroll 8
    for (int i = 0; i < 64; ++i) sum += __expf(sh[i] - mx);
    const float logw = w - (mx + __logf(sum));
    float hld = 0.0f;
    #pragma unroll 8
    for (int i = 0; i < 64; ++i) hld += sr[k * 4096 + i * 64 + i];
    hld *= (1.0f / 512.0f);
    ck_out[k] = logw - hld - 32.0f * LOG_2PI_F;
}

// ---------------------------------------------------------------------------
// Kernel 2: per-component triangular inverse Linv[k] and lm[k] = Linv[k] @ mu[k]
// L = tril(scale_raw/512, -1) + diag(exp(diag(scale_raw/512)))
// One block per component; thread j solves column j by forward substitution.
// ---------------------------------------------------------------------------
__global__ __launch_bounds__(64) void prep_inv(const float* __restrict__ means_raw,
                                               const float* __restrict__ scale_raw,
                                               float* __restrict__ Linv,
                                               float* __restrict__ lmout) {
    __shared__ float Lm[64 * 65];
    __shared__ float Li[64 * 65];
    __shared__ float mu[64];
    const int k = blockIdx.x;
    const int j = threadIdx.x;
    const float* src = scale_raw + k * 4096;
    for (int i = 0; i < 64; ++i) {
        float v = src[i * 64 + j] * (1.0f / 512.0f);
        Lm[i * 65 + j] = (i == j) ? __expf(v) : v;
    }
    mu[j] = means_raw[k * 64 + j] * (1.0f / 64.0f);
    __syncthreads();
    // Column j of Linv (forward substitution), staged through LDS.
    for (int i = 0; i < j; ++i) Li[i * 65 + j] = 0.0f;
    Li[j * 65 + j] = 1.0f / Lm[j * 65 + j];
    for (int i = j + 1; i < 64; ++i) {
        float a = 0.0f;
        for (int m = j; m < i; ++m) a += Lm[i * 65 + m] * Li[m * 65 + j];
        Li[i * 65 + j] = -a / Lm[i * 65 + i];
    }
    __syncthreads();
    // Write Linv row-major (coalesced over j) and lm[j] = row j of Linv dot mu.
    float* dst = Linv + k * 4096;
    for (int i = 0; i < 64; ++i) dst[i * 64 + j] = Li[i * 65 + j];
    float acc = 0.0f;
    for (int m = 0; m <= j; ++m) acc += Li[j * 65 + m] * mu[m];
    lmout[k * 64 + j] = acc;
}

// ---------------------------------------------------------------------------
// Kernel 3: main fused GEMM + streaming logsumexp.
// 256 blocks x 256 threads (8 waves). Wave w handles rows r0..r0+15.
// A (x rows) preloaded into registers once; Linv_k double-buffered in LDS.
// ---------------------------------------------------------------------------
#define LDS_STRIDE 68   // padded row stride in dwords: bank = (4i+off)%64, conflict-free

__global__ __launch_bounds__(256) void gmm_main(const float* __restrict__ x,
                                                const float* __restrict__ Linv,
                                                const float* __restrict__ lm,
                                                const float* __restrict__ ck,
                                                float* __restrict__ part) {
    __shared__ float lbuf[2][64 * LDS_STRIDE];
    __shared__ float red[8];

    const int tid  = threadIdx.x;
    const int wave = tid >> 5;
    const int lane = tid & 31;
    const int half = lane >> 4;      // 0: K parity {0,1}; 1: K parity {2,3}
    const int l16  = lane & 15;
    const int r0   = blockIdx.x * 128 + wave * 16;

    // --- Preload A-fragments: 16 v2f per lane, reused for all 64 components.
    // av[t] = { x[row, 4t + 2*half], x[row, 4t + 2*half + 1] }, row = r0 + l16.
    v2f av[16];
    const float* xrow = x + (size_t)(r0 + l16) * 64 + 2 * half;
    #pragma unroll
    for (int t = 0; t < 16; ++t) av[t] = *(const v2f*)(xrow + 4 * t);

    // --- Stage component 0 into LDS buffer 0 (coalesced b128 loads + b128 ds stores).
    {
        const float* src = Linv;
        #pragma unroll
        for (int c = 0; c < 4; ++c) {
            float4 v = *reinterpret_cast<const float4*>(src + c * 1024 + tid * 4);
            int e = c * 1024 + tid * 4;
            int row = e >> 6, col = e & 63;
            *reinterpret_cast<float4*>(&lbuf[0][row * LDS_STRIDE + col]) = v;
        }
    }
    __syncthreads();

    // Streaming logsumexp state: lanes<16 carry rows r0+0..7, lanes>=16 rows r0+8..15
    // (redundant across the 16-lane group; register-resident, no indexing).
    float m8[8], s8[8];
    #pragma unroll
    for (int r = 0; r < 8; ++r) { m8[r] = -INFINITY; s8[r] = 0.0f; }

    for (int k = 0; k < 64; ++k) {
        const int cur = k & 1, nxt = cur ^ 1;

        // Prefetch k+2 into L2, stage k+1 into registers.
        if (k + 2 < 64) __builtin_prefetch(Linv + (size_t)(k + 2) * 4096, 0, 1);
        float4 st[4];
        if (k + 1 < 64) {
            const float* src = Linv + (size_t)(k + 1) * 4096;
            #pragma unroll
            for (int c = 0; c < 4; ++c)
                st[c] = *reinterpret_cast<const float4*>(src + c * 1024 + tid * 4);
        }

        // Accumulator init C = -lm (so D = Linv@x - lm = z directly).
        const float* lmk = lm + k * 64 + l16;
        v8f acc[4];
        #pragma unroll
        for (int nt = 0; nt < 4; ++nt) {
            float l = -lmk[nt * 16];
            acc[nt] = (v8f){l, l, l, l, l, l, l, l};
        }

        // GEMM: 16 K-steps x 4 N-tiles of v_wmma_f32_16x16x4_f32.
        const float* lbase = &lbuf[cur][0] + 4 * 0 + 2 * half;
        #pragma unroll
        for (int t = 0; t < 16; ++t) {
            const float* bp = &lbuf[cur][0] + 4 * t + 2 * half;
            #pragma unroll
            for (int nt = 0; nt < 4; ++nt) {
                // B[kk, n] = Linv[i0+n][4t + 2*half + kk], lane holds column i = nt*16 + l16
                v2f bv = *(const v2f*)(bp + (nt * 16 + l16) * LDS_STRIDE);
                acc[nt] = __builtin_amdgcn_wmma_f32_16x16x4_f32(
                    false, av[t], false, bv, (short)0, acc[nt], false, false);
            }
        }
        (void)lbase;

        // Store staged k+1 into the other buffer.
        if (k + 1 < 64) {
            #pragma unroll
            for (int c = 0; c < 4; ++c) {
                int e = c * 1024 + tid * 4;
                int row = e >> 6, col = e & 63;
                *reinterpret_cast<float4*>(&lbuf[nxt][row * LDS_STRIDE + col]) = st[c];
            }
        }

        // maha partials: per VGPR r, sum z^2 over this lane's 4 N-tiles.
        float sr[8];
        #pragma unroll
        for (int r = 0; r < 8; ++r) {
            float a0 = acc[0][r], a1 = acc[1][r], a2 = acc[2][r], a3 = acc[3][r];
            sr[r] = a0 * a0 + a1 * a1 + a2 * a2 + a3 * a3;
        }
        // Butterfly reduce across the 16-lane group (N dimension).
        #pragma unroll
        for (int mask = 1; mask <= 8; mask <<= 1) {
            #pragma unroll
            for (int r = 0; r < 8; ++r) sr[r] += __shfl_xor(sr[r], mask, 32);
        }
        // Streaming logsumexp update: v = ck[k] - 0.5 * maha.
        const float ckk = ck[k];
        #pragma unroll
        for (int r = 0; r < 8; ++r) {
            float v  = ckk - 0.5f * sr[r];
            float mn = fmaxf(m8[r], v);
            s8[r] = s8[r] * __expf(m8[r] - mn) + __expf(v - mn);
            m8[r] = mn;
        }
        __syncthreads();   // one barrier per component: flips double buffer
    }

    // Per-row lp = m + log(s); sum the wave's 16 rows; block partial -> ws.
    float lpsum = 0.0f;
    #pragma unroll
    for (int r = 0; r < 8; ++r) lpsum += m8[r] + __logf(s8[r]);
    lpsum += __shfl_xor(lpsum, 16, 32);   // combine rows 0-7 (lanes<16) with 8-15
    if (lane == 0) red[wave] = lpsum;
    __syncthreads();
    if (tid == 0) {
        float t = 0.0f;
        #pragma unroll
        for (int w = 0; w < 8; ++w) t += red[w];
        part[blockIdx.x] = t;
    }
}

// ---------------------------------------------------------------------------
// Kernel 4: deterministic tree reduction of the 256 block partials.
// ---------------------------------------------------------------------------
__global__ __launch_bounds__(256) void finalize_k(const float* __restrict__ part,
                                                  float* __restrict__ out) {
    __shared__ float sh[256];
    const int t = threadIdx.x;
    sh[t] = part[t];
    __syncthreads();
    for (int s = 128; s > 0; s >>= 1) {
        if (t < s) sh[t] += sh[t + s];
        __syncthreads();
    }
    if (t == 0) out[0] = -sh[0] * (1.0f / (float)B_SZ);
}

// ---------------------------------------------------------------------------
extern "C" void kernel_launch(void* const* d_in, const int* in_sizes, int n_in,
                              void* d_out, int out_size, void* d_ws, size_t ws_size,
                              hipStream_t stream) {
    const float* x           = (const float*)d_in[0];   // [B, D]
    const float* means_raw   = (const float*)d_in[1];   // [K, D]
    const float* scale_raw   = (const float*)d_in[2];   // [K, D, D]
    const float* weights_raw = (const float*)d_in[3];   // [K]
    float* out = (float*)d_out;

    // Workspace layout (floats): Linv[64*4096] | lm[64*64] | ck[64] | part[256]
    float* wsf     = (float*)d_ws;
    float* ws_linv = wsf;                 // 262144
    float* ws_lm   = ws_linv + 262144;    // 4096
    float* ws_ck   = ws_lm + 4096;        // 64
    float* ws_part = ws_ck + 64;          // 256

    prep_wh<<<1, 64, 0, stream>>>(weights_raw, scale_raw, ws_ck);
    prep_inv<<<64, 64, 0, stream>>>(means_raw, scale_raw, ws_linv, ws_lm);
    gmm_main<<<256, 256, 0, stream>>>(x, ws_linv, ws_lm, ws_ck, ws_part);
    finalize_k<<<1, 256, 0, stream>>>(ws_part, out);
}